// ParallelFreqAwareEmbeddingBagTablewise_3487513444355
// MI455X (gfx1250) — compile-verified
//
#include <hip/hip_runtime.h>

// ---------------------------------------------------------------------------
// Table-wise EmbeddingBag (mean) for MI455X / gfx1250.
//
// Shapes (from reference): T=8 tables, D=64, bags = T*B = 131072, L = 20
// ids per bag, weight = [T*V, D] fp32 (512 MB), out = [B, T*D] fp32.
//
// Roofline: ~700 MB moved per call vs ~0.17 GFLOP -> pure gather-bandwidth
// bound (~30 us at 23.3 TB/s). Strategy: one wave32 per bag, lane = float2
// slice of the 64-wide row, so each gathered row is one wave-coalesced 256B
// async B64 transfer into LDS (ASYNCcnt path); 20 transfers issued
// back-to-back (fully unrolled), one s_wait_asynccnt, then reduce from LDS
// (backend pairs the reads into ds_load_2addr_b64). Output is stored with
// TH=NT so the 32 MB result stream does not evict weight lines from the
// 192 MB L2 (weight table is the only data with reuse value).
// ---------------------------------------------------------------------------

#if defined(__HIP_DEVICE_COMPILE__) && \
    __has_builtin(__builtin_amdgcn_global_load_async_to_lds_b64) && \
    __has_builtin(__builtin_amdgcn_s_wait_asynccnt)
#define EMB_ASYNC 1
#else
#define EMB_ASYNC 0
#endif

// The async-to-LDS b64 builtin takes pointers to 8-byte int vectors:
//   (v2i addrspace(1)* src, v2i addrspace(3)* dst, imm offset, imm cpol)
typedef int v2i __attribute__((ext_vector_type(2)));
typedef __attribute__((address_space(1))) v2i* GlobV2iPtr;  // global (AS1)
typedef __attribute__((address_space(3))) v2i* LdsV2iPtr;   // LDS (AS3)

__device__ __forceinline__ void gather_row_b64(const float* __restrict__ g,
                                               float* l) {
#if EMB_ASYNC
  // Per-lane 8B: lane k copies weight row bytes [8k, 8k+8) -> LDS. One wave
  // instruction moves a full 256B embedding row. Tracked by ASYNCcnt.
  __builtin_amdgcn_global_load_async_to_lds_b64((GlobV2iPtr)g, (LdsV2iPtr)l,
                                                /*imm offset=*/0,
                                                /*cpol=*/0);
#else
  *(float2*)l = *(const float2*)g;
#endif
}

__device__ __forceinline__ void wait_gather_done() {
#if EMB_ASYNC
  __builtin_amdgcn_s_wait_asynccnt(0);
  asm volatile("" ::: "memory");  // keep LDS reads below the wait
#endif
}

namespace {
constexpr int D     = 64;   // embedding dim (2 floats per lane, wave32)
constexpr int WAVES = 8;    // waves (bags) per block
constexpr int CHUNK = 20;   // ids per bag in this workload (fast path)
}

__global__ void __launch_bounds__(WAVES * 32)
embedding_bag_mean_kernel(const int* __restrict__ indices,
                          const int* __restrict__ offsets,
                          const float* __restrict__ weight,
                          float* __restrict__ out,
                          int N,        // total indices
                          int numBags,  // T * B
                          int B)        // batch (bags per table)
{
  extern __shared__ __align__(16) float smem[];  // WAVES * CHUNK * D floats

  const int lane = threadIdx.x & 31;
  // bag is wave-uniform by construction; readfirstlane makes that visible to
  // the compiler -> offsets/indices addressing goes scalar (s_load + saddr).
  const int bag = __builtin_amdgcn_readfirstlane(
      (int)blockIdx.x * WAVES + (int)(threadIdx.x >> 5));
  if (bag >= numBags) return;

  float* stage = smem + (size_t)(threadIdx.x >> 5) * (CHUNK * D);

  const int start = offsets[bag];
  const int end   = (bag + 1 < numBags) ? offsets[bag + 1] : N;
  const int count = end - start;

  // 4 independent accumulators to break the FP-add dependence chain.
  float a0x = 0.f, a0y = 0.f, a1x = 0.f, a1y = 0.f;
  float a2x = 0.f, a2y = 0.f, a3x = 0.f, a3y = 0.f;

  if (count == CHUNK) {
    // -------- fast path: fully unrolled, 20 async gathers in flight --------
    int myIdx = (lane < CHUNK) ? indices[start + lane] : 0;  // one coalesced load

#pragma unroll
    for (int j = 0; j < CHUNK; ++j) {
      const int idx = __builtin_amdgcn_readlane(myIdx, j);   // imm lane -> SGPR
      gather_row_b64(weight + (size_t)idx * D + lane * 2,
                     stage + j * D + lane * 2);
    }

    wait_gather_done();

#pragma unroll
    for (int j = 0; j < CHUNK; j += 4) {
      float2 v0 = *(const float2*)(stage + (j + 0) * D + lane * 2);
      float2 v1 = *(const float2*)(stage + (j + 1) * D + lane * 2);
      float2 v2 = *(const float2*)(stage + (j + 2) * D + lane * 2);
      float2 v3 = *(const float2*)(stage + (j + 3) * D + lane * 2);
      a0x += v0.x; a0y += v0.y;
      a1x += v1.x; a1y += v1.y;
      a2x += v2.x; a2y += v2.y;
      a3x += v3.x; a3y += v3.y;
    }
  } else {
    // -------- generic path: chunk by CHUNK, dynamic trip counts --------
    for (int base = 0; base < count; base += CHUNK) {
      int c = count - base;
      if (c > CHUNK) c = CHUNK;

      int myIdx = (lane < c) ? indices[start + base + lane] : 0;

      for (int j = 0; j < c; ++j) {
        const int idx = __builtin_amdgcn_readlane(myIdx, j);
        gather_row_b64(weight + (size_t)idx * D + lane * 2,
                       stage + j * D + lane * 2);
      }

      wait_gather_done();

      int j = 0;
      for (; j + 3 < c; j += 4) {
        float2 v0 = *(const float2*)(stage + (j + 0) * D + lane * 2);
        float2 v1 = *(const float2*)(stage + (j + 1) * D + lane * 2);
        float2 v2 = *(const float2*)(stage + (j + 2) * D + lane * 2);
        float2 v3 = *(const float2*)(stage + (j + 3) * D + lane * 2);
        a0x += v0.x; a0y += v0.y;
        a1x += v1.x; a1y += v1.y;
        a2x += v2.x; a2y += v2.y;
        a3x += v3.x; a3y += v3.y;
      }
      for (; j < c; ++j) {
        float2 v = *(const float2*)(stage + j * D + lane * 2);
        a0x += v.x; a0y += v.y;
      }
      // Accumulator consumption (dscnt waits) precedes the next chunk's async
      // issues in program order, so LDS reuse is safe.
    }
  }

  const float cnt = (float)((count > 1) ? count : 1);
  float2 r;
  r.x = ((a0x + a1x) + (a2x + a3x)) / cnt;  // matches reference sums/max(cnt,1)
  r.y = ((a0y + a1y) + (a2y + a3y)) / cnt;

  // Bags are (table, batch) ordered; output is [B, T*D] with per-table concat.
  const int t = bag / B;
  const int b = bag - t * B;
  const int T = numBags / B;
  float* o = out + (size_t)b * ((size_t)T * D) + (size_t)t * D + lane * 2;

  // Non-temporal b64 store: output has no reuse, keep it out of L2 so the
  // weight table retains more residency.
  v2i bits;
  __builtin_memcpy(&bits, &r, sizeof(bits));
  __builtin_nontemporal_store(bits, (v2i*)o);
}

extern "C" void kernel_launch(void* const* d_in, const int* in_sizes, int n_in,
                              void* d_out, int out_size, void* d_ws, size_t ws_size,
                              hipStream_t stream) {
  (void)n_in; (void)d_ws; (void)ws_size; (void)out_size;

  const int*   indices = (const int*)d_in[0];
  const int*   offsets = (const int*)d_in[1];
  const float* weight  = (const float*)d_in[2];
  float*       out     = (float*)d_out;

  const int N       = in_sizes[0];   // T*B*L = 2,621,440
  const int numBags = in_sizes[1];   // T*B   = 131,072
  constexpr int T   = 8;             // matches reference constant
  const int B       = numBags / T;   // 16,384

  const int blocks = (numBags + WAVES - 1) / WAVES;
  const size_t ldsBytes = (size_t)WAVES * CHUNK * D * sizeof(float);  // 40 KB

  embedding_bag_mean_kernel<<<blocks, WAVES * 32, ldsBytes, stream>>>(
      indices, offsets, weight, out, N, numBags, B);
}